// ktree_synapse_62749472194921
// MI455X (gfx1250) — compile-verified
//
#include <hip/hip_runtime.h>

// ---------------- problem constants ----------------
#define BTOT   8192
#define K0     3072
#define ROWS   32          // rows (b) per block == lanes per wave
#define WAVES  4           // k-split: waves per block
#define LPW    768         // leaves per wave (3072/4)
#define CHUNK  48          // leaves per chunk (complete subtree -> one K=64 node)
#define NCHUNK 16          // chunks per wave
#define PSTR   52          // padded dwords per row in LDS tile (208B, 16B aligned)
#define CHBYTES (ROWS * PSTR * 4)   // 6656 B per chunk buffer

typedef __attribute__((ext_vector_type(2))) float v2f;
typedef __attribute__((ext_vector_type(8))) float v8f;

struct Params {
    const float* x;
    const float* ap1; const float* ap2;
    const float* am1; const float* am2;
    const float* gz0;
    const float* W[11];   // w0:(R,1024,3), w1:(R,512,2), ... w10:(R,1,2)
    const float* Bb[11];  // b0:(R,1024), ...
    const float* rootw;   // (1,R)
    const float* rootb;   // (1,)
    float* f0;            // ws: feats r=0  (BTOT)
    float* f1;            // ws: feats r=1  (BTOT)
    float* out;           // (BTOT)
    int direct;           // 1 -> kernel1 writes out directly (ws too small)
};

__device__ __forceinline__ float leaky_(float v) {
    return fmaxf(v, 0.0f) + 0.01f * fminf(v, 0.0f);
}

// combine two child nodes a,b -> parent node m at a level whose weight array
// has shape (R, KO, 2) and bias (R, KO)
__device__ __forceinline__ float fold2(const float* W, const float* Bp,
                                       int r, int KO, int m, float a, float b) {
    const int o = (r * KO + m) * 2;
    float v = fmaf(W[o], a, fmaf(W[o + 1], b, Bp[r * KO + m]));
    return leaky_(v);
}

// synapse: (gp*50 + gm*(-70) + gz*(-65)) / (gp + gm + gz)
__device__ __forceinline__ float synapse(const Params& p, int r, int k, float xv) {
    const int o = r * K0 + k;
    float tp = __builtin_amdgcn_fmed3f(fmaf(p.ap1[o], xv, p.ap2[o]), -60.0f, 60.0f);
    float tm = __builtin_amdgcn_fmed3f(fmaf(p.am1[o], xv, p.am2[o]), -60.0f, 60.0f);
    float gp = __expf(tp);                 // v_exp_f32 path
    float gm = __expf(tm);
    float gz = p.gz0[o];                   // note: reference applies NO exp to g0
    float num = fmaf(gp, 50.0f, fmaf(gm, -70.0f, gz * -65.0f));
    float den = gp + gm + gz;
    return num * __builtin_amdgcn_rcpf(den);
}

// async global -> LDS copy of one chunk: 32 rows x 48 floats (384 x b128)
__device__ __forceinline__ void load_chunk_async(const float* x, int b0, int kb,
                                                 unsigned ldsbase_bytes, int lane) {
#pragma unroll
    for (int t = 0; t < 12; ++t) {
        int idx = t * 32 + lane;           // 0..383
        int row = idx / 12;
        int c   = idx % 12;                // b128 index within row
        const float* g = x + (size_t)(b0 + row) * K0 + kb + c * 4;
        unsigned l = ldsbase_bytes + (unsigned)(row * (PSTR * 4) + c * 16);
        asm volatile("global_load_async_to_lds_b128 %0, %1, off"
                     :: "v"(l), "v"(g) : "memory");
    }
}

__global__ __launch_bounds__(ROWS * WAVES)
void ktree_main(Params p) {
    __shared__ __align__(16) float xs[WAVES * 2 * ROWS * PSTR]; // 53,248 B
    __shared__ float part[2][WAVES][ROWS];                      // 1 KB

    const int lane = threadIdx.x & 31;
    const int wave = __builtin_amdgcn_readfirstlane((int)(threadIdx.x >> 5)); // force SGPR
    const int b0   = blockIdx.x * ROWS;
    const int kw   = wave * LPW;

    const unsigned wave_base =
        (unsigned)(size_t)(void*)xs + (unsigned)(wave * 2 * CHBYTES);

    // prefetch chunk 0
    load_chunk_async(p.x, b0, kw, wave_base, lane);

    float cst0[4], cst1[4];
    float v9_0 = 0.0f, v9_1 = 0.0f;

#pragma unroll 1
    for (int c = 0; c < NCHUNK; ++c) {
        // prefetch next chunk into the other buffer, then wait for current one.
        if (c + 1 < NCHUNK) {
            load_chunk_async(p.x, b0, kw + (c + 1) * CHUNK,
                             wave_base + (unsigned)(((c + 1) & 1) * CHBYTES), lane);
            asm volatile("s_wait_asynccnt 0xc" ::: "memory"); // 12 outstanding = next chunk only
        } else {
            asm volatile("s_wait_asynccnt 0x0" ::: "memory");
        }

        const float* xb = &xs[(wave * 2 + (c & 1)) * (ROWS * PSTR) + lane * PSTR];

        const int kb  = kw + c * CHUNK;    // leaf base of this chunk (uniform)
        const int n1b = kb / 3;            // first level-1 node index (uniform)

        float stk0[4], stk1[4];
        float v0 = 0.0f, v1 = 0.0f;

        // 16 level-1 nodes (groups of 3 leaves), folded in-chunk down to one K=64 node
#pragma unroll
        for (int g = 0; g < 16; ++g) {
            float s0 = 0.0f, s1 = 0.0f;
            const int n1 = n1b + g;
#pragma unroll
            for (int t = 0; t < 3; ++t) {
                const float xv = xb[g * 3 + t];       // ds_load_b32 (padded stride)
                const int   k  = kb + g * 3 + t;      // uniform -> scalar param loads
                s0 = fmaf(p.W[0][n1 * 3 + t],          synapse(p, 0, k, xv), s0);
                s1 = fmaf(p.W[0][K0 + n1 * 3 + t],     synapse(p, 1, k, xv), s1);
            }
            v0 = leaky_(s0 + p.Bb[0][n1]);
            v1 = leaky_(s1 + p.Bb[0][1024 + n1]);

            // static stack fold: levels using w1..w4 (KO = 512,256,128,64)
#pragma unroll
            for (int l = 0; l < 4; ++l) {
                if (g & (1 << l)) {
                    const int m  = n1 >> (l + 1);
                    const int KO = 512 >> l;
                    v0 = fold2(p.W[l + 1], p.Bb[l + 1], 0, KO, m, stk0[l], v0);
                    v1 = fold2(p.W[l + 1], p.Bb[l + 1], 1, KO, m, stk1[l], v1);
                } else { stk0[l] = v0; stk1[l] = v1; break; }
            }
        }

        // cross-chunk fold: levels using w5..w8 (KO = 32,16,8,4); node = wave*16 + c
        const int n5 = wave * NCHUNK + c;
#pragma unroll
        for (int l = 0; l < 4; ++l) {
            if (c & (1 << l)) {
                const int m  = n5 >> (l + 1);
                const int KO = 32 >> l;
                v0 = fold2(p.W[5 + l], p.Bb[5 + l], 0, KO, m, cst0[l], v0);
                v1 = fold2(p.W[5 + l], p.Bb[5 + l], 1, KO, m, cst1[l], v1);
            } else { cst0[l] = v0; cst1[l] = v1; break; }
        }
        if (c == NCHUNK - 1) { v9_0 = v0; v9_1 = v1; }
    }

    // per-wave K=4 node -> LDS, combine in wave 0
    part[0][wave][lane] = v9_0;
    part[1][wave][lane] = v9_1;
    __syncthreads();

    if (threadIdx.x < 32) {
        const int b = b0 + lane;
        float y[2];
#pragma unroll
        for (int r = 0; r < 2; ++r) {
            float q0 = fold2(p.W[9],  p.Bb[9],  r, 2, 0, part[r][0][lane], part[r][1][lane]);
            float q1 = fold2(p.W[9],  p.Bb[9],  r, 2, 1, part[r][2][lane], part[r][3][lane]);
            y[r]     = fold2(p.W[10], p.Bb[10], r, 1, 0, q0, q1);
        }
        if (p.direct) {
            p.out[b] = fmaf(y[0], p.rootw[0], fmaf(y[1], p.rootw[1], p.rootb[0]));
        } else {
            p.f0[b] = y[0];
            p.f1[b] = y[1];
        }
    }
}

// ---------- root GEMM: out = feats(B,2) @ root_w.T(2,1) + root_b ----------
// One V_WMMA_F32_16X16X4_F32 per 16-row tile. A: M=row, K = [y0,y1,0,0];
// B column 0 = [rw0, rw1, 0, 0]. D column N=0 lives in lane 0 (M=0..7) and
// lane 16 (M=8..15) per the 16x16 f32 C/D layout.
__global__ __launch_bounds__(256)
void ktree_root_wmma(const float* __restrict__ f0, const float* __restrict__ f1,
                     const float* __restrict__ rw, const float* __restrict__ rb,
                     float* __restrict__ out) {
    const int lane = threadIdx.x & 31;
    const int tile = blockIdx.x * (blockDim.x >> 5) + (threadIdx.x >> 5);
    const int mrow = tile * 16 + (lane & 15);

    // A layout (16x4 f32): VGPR0 = K0 (lanes 0-15) / K2 (lanes 16-31),
    //                      VGPR1 = K1 / K3.  K2,K3 zero-padded.
    v2f a;
    a.x = (lane < 16) ? f0[mrow] : 0.0f;
    a.y = (lane < 16) ? f1[mrow] : 0.0f;
    // B (4x16 f32): only column N=0 nonzero -> lane 0 holds rw0 (K0) / rw1 (K1)
    v2f b;
    b.x = (lane == 0) ? rw[0] : 0.0f;
    b.y = (lane == 0) ? rw[1] : 0.0f;

    v8f acc = {};
    acc = __builtin_amdgcn_wmma_f32_16x16x4_f32(false, a, false, b,
                                                (short)0, acc, false, false);
    const float bias = rb[0];
    if (lane == 0) {
#pragma unroll
        for (int j = 0; j < 8; ++j) out[tile * 16 + j] = acc[j] + bias;
    } else if (lane == 16) {
#pragma unroll
        for (int j = 0; j < 8; ++j) out[tile * 16 + 8 + j] = acc[j] + bias;
    }
}

extern "C" void kernel_launch(void* const* d_in, const int* in_sizes, int n_in,
                              void* d_out, int out_size, void* d_ws, size_t ws_size,
                              hipStream_t stream) {
    Params p;
    p.x   = (const float*)d_in[0];
    p.ap1 = (const float*)d_in[1];
    p.ap2 = (const float*)d_in[2];
    p.am1 = (const float*)d_in[3];
    p.am2 = (const float*)d_in[4];
    p.gz0 = (const float*)d_in[5];
    // dict order interleaves w_i, b_i
    for (int i = 0; i < 11; ++i) {
        p.W[i]  = (const float*)d_in[6 + 2 * i];
        p.Bb[i] = (const float*)d_in[7 + 2 * i];
    }
    p.rootw = (const float*)d_in[28];
    p.rootb = (const float*)d_in[29];

    const bool use_ws = ws_size >= (size_t)(2 * BTOT * sizeof(float));
    p.f0 = (float*)d_ws;
    p.f1 = p.f0 + BTOT;
    p.out = (float*)d_out;
    p.direct = use_ws ? 0 : 1;

    ktree_main<<<BTOT / ROWS, ROWS * WAVES, 0, stream>>>(p);

    if (use_ws) {
        // 8192/16 = 512 tiles, 8 waves (tiles) per 256-thread block
        ktree_root_wmma<<<(BTOT / 16) / 8, 256, 0, stream>>>(p.f0, p.f1, p.rootw,
                                                             p.rootb, (float*)d_out);
    }
}